// RobustWarpLoss_10531259810001
// MI455X (gfx1250) — compile-verified
//
#include <hip/hip_runtime.h>
#include <hip/hip_bf16.h>
#include <cstdint>

#define BLOCK   256
#define NLAB    14          // C
#define MAX_T   13          // C - 1
#define L_PARAM 14
#define KAPPA   2.0f
#define RAMP_CAP 1.8f

#if defined(__HIP_DEVICE_COMPILE__)
// gfx1250 async-to-LDS builtin signature (confirmed via clang diagnostic):
//   (AS1 int* global_src, AS3 int* lds_dst, Imm int offset, Imm int cpol)
typedef __attribute__((address_space(1))) int glb_i32_t;
typedef __attribute__((address_space(3))) int lds_i32_t;
#endif

__device__ __forceinline__ float ramp_clip(float v) {
    return fminf(fmaxf(v, 0.0f), RAMP_CAP);
}

// ---------------------------------------------------------------------------
// Kernel 1: one thread per row. Stage inp/target rows through LDS with
// coalesced async-to-LDS loads (ASYNCcnt path), stream rand_u lazily with
// non-temporal loads + WGP-scope prefetch, emit one partial sum per block.
// ---------------------------------------------------------------------------
__global__ void __launch_bounds__(BLOCK)
warp_loss_partial(const float* __restrict__ inp,
                  const int*   __restrict__ tgt,
                  const float* __restrict__ S,
                  const float* __restrict__ rand_u,
                  float* __restrict__ partial,
                  int B)
{
    __shared__ float s_inp[BLOCK * NLAB];   // 14336 B
    __shared__ int   s_tgt[BLOCK * NLAB];   // 14336 B
    __shared__ float s_L[NLAB];             // harmonic cumsum table
    __shared__ float s_red[BLOCK];          //  1024 B

    const int tid  = threadIdx.x;
    const int row0 = blockIdx.x * BLOCK;
    const int i    = row0 + tid;

    // Early prefetch of this thread's rand_u region (728 B): locality 3 ->
    // SCOPE_WGP, i.e. pull into ALL cache levels while we stage/build masks.
    if (i < B) {
        const char* rpre = (const char*)(rand_u + (size_t)i * (NLAB * MAX_T));
        __builtin_prefetch(rpre,       0, 3);
        __builtin_prefetch(rpre + 256, 0, 3);
        __builtin_prefetch(rpre + 512, 0, 3);
    }

    // L[r] = cumsum_{m=1..r+1} 1/m, accumulated in f32 in reference order.
    if (tid == 0) {
        float h = 0.0f;
#pragma unroll
        for (int r = 0; r < NLAB; ++r) {
            h += 1.0f / (float)(r + 1);
            s_L[r] = h;
        }
    }

    // ---- coalesced staging of 256 rows of inp/target into LDS ----
    const int    rows  = min(BLOCK, B - row0);
    const size_t gbase = (size_t)row0 * NLAB;
    const int    nelem = rows * NLAB;

#if defined(__HIP_DEVICE_COMPILE__) && \
    __has_builtin(__builtin_amdgcn_global_load_async_to_lds_b32) && \
    __has_builtin(__builtin_amdgcn_s_wait_asynccnt)
    for (int e = tid; e < nelem; e += BLOCK) {
        __builtin_amdgcn_global_load_async_to_lds_b32(
            (glb_i32_t*)(void*)(inp + gbase + e),
            (lds_i32_t*)(void*)&s_inp[e], 0, 0);
        __builtin_amdgcn_global_load_async_to_lds_b32(
            (glb_i32_t*)(void*)(tgt + gbase + e),
            (lds_i32_t*)(void*)&s_tgt[e], 0, 0);
    }
    __builtin_amdgcn_s_wait_asynccnt(0);
#else
    for (int e = tid; e < nelem; e += BLOCK) {
        s_inp[e] = __builtin_nontemporal_load(inp + gbase + e);
        s_tgt[e] = __builtin_nontemporal_load(tgt + gbase + e);
    }
#endif
    __syncthreads();

    float acc = 0.0f;
    if (i < B) {
        // pull my row into registers
        float x[NLAB];
        int   tg[NLAB];
#pragma unroll
        for (int c = 0; c < NLAB; ++c) {
            x[c]  = s_inp[tid * NLAB + c];
            tg[c] = s_tgt[tid * NLAB + c];
        }

        // pack negative-label indices (stable order), 4 bits each
        uint64_t negpack = 0;
        int nneg = 0;
#pragma unroll
        for (int c = 0; c < NLAB; ++c) {
            if (tg[c] == 0) {
                negpack |= ((uint64_t)c) << (4 * nneg);
                ++nneg;
            }
        }

        // kappa term: clip(1 - t*x, 0, 1.8) over all labels
        float kap = 0.0f;
#pragma unroll
        for (int c = 0; c < NLAB; ++c) {
            kap += ramp_clip(1.0f - (float)tg[c] * x[c]);
        }

        // WARP term: only positives with at least one negative
        float warp = 0.0f;
        if (nneg > 0) {
            const float* rbase = rand_u + (size_t)i * (NLAB * MAX_T);
            const float  fn    = (float)nneg;
#pragma unroll
            for (int j = 0; j < NLAB; ++j) {
                if (tg[j] != 1) continue;

                // neg_sum[j] = sum over negatives k of clip(x_j - x_k, 0, 1.8)
                float ns = 0.0f;
#pragma unroll
                for (int c = 0; c < NLAB; ++c) {
                    if (tg[c] == 0) ns += ramp_clip(x[j] - x[c]);
                }

                // sequential trials: stop at first margin >= 0 (cap MAX_T)
                int nt = MAX_T;
                for (int t = 0; t < MAX_T; ++t) {
                    float u  = __builtin_nontemporal_load(rbase + j * MAX_T + t);
                    int   ti = (int)(u * fn);
                    ti = max(0, min(ti, nneg - 1));
                    int   k  = (int)((negpack >> (4 * ti)) & 0xFull);
                    float xv = s_inp[tid * NLAB + k];   // dynamic idx via LDS
                    if (1.0f - x[j] + xv >= 0.0f) { nt = t + 1; break; }
                }
                warp += s_L[MAX_T / nt] * ns;           // weight = L[r_j]
            }
        }

        const float s_i = (i <= L_PARAM) ? 1.0f : S[i - L_PARAM];
        acc = s_i * (warp + KAPPA * kap);
    }

    // deterministic block tree reduction
    s_red[tid] = acc;
    __syncthreads();
#pragma unroll
    for (int s = BLOCK / 2; s > 0; s >>= 1) {
        if (tid < s) s_red[tid] += s_red[tid + s];
        __syncthreads();
    }
    if (tid == 0) partial[blockIdx.x] = s_red[0];
}

// ---------------------------------------------------------------------------
// Kernel 2: single-block deterministic reduction of per-block partials.
// ---------------------------------------------------------------------------
__global__ void __launch_bounds__(BLOCK)
warp_loss_finalize(const float* __restrict__ partial, int n,
                   float* __restrict__ out)
{
    __shared__ float s_red[BLOCK];
    float a = 0.0f;
    for (int idx = threadIdx.x; idx < n; idx += BLOCK) a += partial[idx];
    s_red[threadIdx.x] = a;
    __syncthreads();
#pragma unroll
    for (int s = BLOCK / 2; s > 0; s >>= 1) {
        if (threadIdx.x < s) s_red[threadIdx.x] += s_red[threadIdx.x + s];
        __syncthreads();
    }
    if (threadIdx.x == 0) out[0] = s_red[0];
}

// ---------------------------------------------------------------------------
// Host launcher. Inputs (setup_inputs order): inp [B,C] f32, target [B,C] i32,
// S [B] f32, rand_u [B,C,MAX_T] f32. Output: scalar f32.
// ---------------------------------------------------------------------------
extern "C" void kernel_launch(void* const* d_in, const int* in_sizes, int n_in,
                              void* d_out, int out_size, void* d_ws, size_t ws_size,
                              hipStream_t stream)
{
    const float* inp    = (const float*)d_in[0];
    const int*   target = (const int*)d_in[1];
    const float* S      = (const float*)d_in[2];
    const float* rand_u = (const float*)d_in[3];
    float*       out    = (float*)d_out;

    const int B       = in_sizes[2];                 // S has B elements
    const int nblocks = (B + BLOCK - 1) / BLOCK;     // 1024 for B = 262144

    float* partial = (float*)d_ws;                   // nblocks floats of scratch

    warp_loss_partial<<<nblocks, BLOCK, 0, stream>>>(inp, target, S, rand_u,
                                                     partial, B);
    warp_loss_finalize<<<1, BLOCK, 0, stream>>>(partial, nblocks, out);

    (void)n_in; (void)out_size; (void)ws_size;
}